// BatchUnProcessDatav2_85469849191057
// MI455X (gfx1250) — compile-verified
//
#include <hip/hip_runtime.h>
#include <hip/hip_bf16.h>

// Vector/pointer types for the gfx1250 async global->LDS builtin, whose
// params are pointers to 4xi32 vectors (per hipcc diagnostic).
typedef int v4i_async __attribute__((vector_size(16)));
typedef __attribute__((address_space(1))) v4i_async g_v4i;
typedef __attribute__((address_space(3))) v4i_async l_v4i;

// ---------------------------------------------------------------------------
// Quaternion helpers (rotation by q with components already holding conj(q)).
// rotate(q, p): t = 2*(qv x p); p' = p + w*t + qv x t
// ---------------------------------------------------------------------------

__device__ __forceinline__ void quat_rotate(float qw, float qx, float qy, float qz,
                                            float px, float py, float pz,
                                            float& rx, float& ry, float& rz) {
    float cx = qy * pz - qz * py;
    float cy = qz * px - qx * pz;
    float cz = qx * py - qy * px;
    cx += cx; cy += cy; cz += cz;
    rx = px + qw * cx + (qy * cz - qz * cy);
    ry = py + qw * cy + (qz * cx - qx * cz);
    rz = pz + qw * cz + (qx * cy - qy * cx);
}

// ---------------------------------------------------------------------------
// Kernel 1: per-batch trajectory scan.
// traj[b,t] (x,z only) = rot(inv(root[b,0]), pos[b,0,0]) + sum_{s<t} rot(inv(root[b,s]), vel[b,s,0])
// One block per batch, T threads, two-level wave32 shuffle scan.
// ---------------------------------------------------------------------------
__global__ __launch_bounds__(1024) void traj_scan_kernel(
    const float* __restrict__ glb_pos, const float* __restrict__ glb_vel,
    const float* __restrict__ root, float* __restrict__ traj, int T, int J) {
    const int b    = blockIdx.x;
    const int t    = threadIdx.x;
    const int lane = t & 31;
    const int wave = t >> 5;

    __shared__ float s_sum[64];  // 32 waves * 2 channels (inclusive wave totals)
    __shared__ float s_off[64];  // inclusive scan of wave totals

    float wx = 0.0f, wz = 0.0f;
    if (t > 0) {
        const float* q = root + ((size_t)b * T + (t - 1)) * 4;
        float qw = q[0], qx = -q[1], qy = -q[2], qz = -q[3];
        const float* v = glb_vel + ((size_t)b * (T - 1) + (t - 1)) * (size_t)J * 3;
        float rx, ry, rz;
        quat_rotate(qw, qx, qy, qz, v[0], v[1], v[2], rx, ry, rz);
        wx = rx; wz = rz;  // y channel is masked out downstream
    }

    // intra-wave inclusive scan (wave32)
    float sx = wx, sz = wz;
    #pragma unroll
    for (int d = 1; d < 32; d <<= 1) {
        float ox = __shfl_up(sx, d, 32);
        float oz = __shfl_up(sz, d, 32);
        if (lane >= d) { sx += ox; sz += oz; }
    }
    if (lane == 31) { s_sum[wave * 2] = sx; s_sum[wave * 2 + 1] = sz; }
    __syncthreads();

    const int nw = blockDim.x >> 5;
    if (wave == 0) {
        float ax = (lane < nw) ? s_sum[lane * 2]     : 0.0f;
        float az = (lane < nw) ? s_sum[lane * 2 + 1] : 0.0f;
        #pragma unroll
        for (int d = 1; d < 32; d <<= 1) {
            float ox = __shfl_up(ax, d, 32);
            float oz = __shfl_up(az, d, 32);
            if (lane >= d) { ax += ox; az += oz; }
        }
        s_off[lane * 2] = ax; s_off[lane * 2 + 1] = az;
    }
    __syncthreads();
    if (wave > 0) { sx += s_off[(wave - 1) * 2]; sz += s_off[(wave - 1) * 2 + 1]; }

    // base term: joint 0 at time 0 rotated into root frame (broadcast loads, cache-served)
    const float* q0 = root + (size_t)b * T * 4;
    float qw = q0[0], qx = -q0[1], qy = -q0[2], qz = -q0[3];
    const float* p0 = glb_pos + (size_t)b * T * (size_t)J * 3;
    float bx, by, bz;
    quat_rotate(qw, qx, qy, qz, p0[0], p0[1], p0[2], bx, by, bz);

    traj[((size_t)b * T + t) * 2]     = bx + sx;
    traj[((size_t)b * T + t) * 2 + 1] = bz + sz;
}

// ---------------------------------------------------------------------------
// Kernel 2 (J == 32 fast path): one wave per (b,t), lane = joint.
// 8 (b,t) pairs per 256-thread block. Root quaternions staged global->LDS
// with CDNA5 async-to-LDS; trajectory staged alongside.
// ---------------------------------------------------------------------------
#define APPLY_PAIRS 8

__global__ __launch_bounds__(256) void apply32_kernel(
    const float* __restrict__ glb_pos, const float* __restrict__ glb_rot,
    const float* __restrict__ root, const float* __restrict__ traj,
    float* __restrict__ out_pos, float* __restrict__ out_rot) {
    __shared__ float s_q[APPLY_PAIRS * 4];
    __shared__ float s_t[APPLY_PAIRS * 2];

    const int tid      = threadIdx.x;
    const int pairBase = blockIdx.x * APPLY_PAIRS;

    if (tid < APPLY_PAIRS) {
        const float* gq = root + ((size_t)pairBase + tid) * 4;
#if defined(__gfx1250__) && __has_builtin(__builtin_amdgcn_global_load_async_to_lds_b128)
        __builtin_amdgcn_global_load_async_to_lds_b128(
            (g_v4i*)(void*)gq,
            (l_v4i*)(void*)(&s_q[tid * 4]),
            /*offset=*/0, /*cpol=*/0);
#else
        unsigned lds_off = (unsigned)(unsigned long long)
            (__attribute__((address_space(3))) float*)(&s_q[tid * 4]);
        asm volatile("global_load_async_to_lds_b128 %0, %1, off"
                     :: "v"(lds_off), "v"(gq) : "memory");
#endif
    }
    if (tid < APPLY_PAIRS * 2) s_t[tid] = traj[(size_t)pairBase * 2 + tid];

#if defined(__gfx1250__) && __has_builtin(__builtin_amdgcn_s_wait_asynccnt)
    __builtin_amdgcn_s_wait_asynccnt(0);
#else
    asm volatile("s_wait_asynccnt 0" ::: "memory");
#endif
    __syncthreads();

    const int wq   = tid >> 5;          // which (b,t) pair in this block
    const int lane = tid & 31;          // joint index
    const size_t p   = (size_t)pairBase + wq;
    const size_t idx = p * 32 + lane;   // flat (b,t,j)

    const float qw = s_q[wq * 4 + 0];
    const float qx = -s_q[wq * 4 + 1];
    const float qy = -s_q[wq * 4 + 2];
    const float qz = -s_q[wq * 4 + 3];

    // positions: rotate then add masked trajectory (x,z only)
    const float* pp = glb_pos + idx * 3;
    float rx, ry, rz;
    quat_rotate(qw, qx, qy, qz, pp[0], pp[1], pp[2], rx, ry, rz);
    float* po = out_pos + idx * 3;
    po[0] = rx + s_t[wq * 2 + 0];
    po[1] = ry;
    po[2] = rz + s_t[wq * 2 + 1];

    // rotations: quat_mul(inv, r), standardized; b128 load/store
    const float4 r = *(const float4*)(glb_rot + idx * 4);
    float ow = qw * r.x - qx * r.y - qy * r.z - qz * r.w;
    float ox = qw * r.y + qx * r.x + qy * r.w - qz * r.z;
    float oy = qw * r.z - qx * r.w + qy * r.x + qz * r.y;
    float oz = qw * r.w + qx * r.z - qy * r.y + qz * r.x;
    if (ow < 0.0f) { ow = -ow; ox = -ox; oy = -oy; oz = -oz; }
    float4 o; o.x = ow; o.y = ox; o.z = oy; o.w = oz;
    *(float4*)(out_rot + idx * 4) = o;
}

// ---------------------------------------------------------------------------
// Generic fallback (any J): one thread per (b,t,j), no LDS staging.
// ---------------------------------------------------------------------------
__global__ void apply_generic_kernel(
    const float* __restrict__ glb_pos, const float* __restrict__ glb_rot,
    const float* __restrict__ root, const float* __restrict__ traj,
    float* __restrict__ out_pos, float* __restrict__ out_rot,
    long long total, int J) {
    long long idx = (long long)blockIdx.x * blockDim.x + threadIdx.x;
    if (idx >= total) return;
    long long p = idx / J;

    const float* q = root + p * 4;
    const float qw = q[0], qx = -q[1], qy = -q[2], qz = -q[3];

    const float* pp = glb_pos + idx * 3;
    float rx, ry, rz;
    quat_rotate(qw, qx, qy, qz, pp[0], pp[1], pp[2], rx, ry, rz);
    float* po = out_pos + idx * 3;
    po[0] = rx + traj[p * 2 + 0];
    po[1] = ry;
    po[2] = rz + traj[p * 2 + 1];

    const float4 r = *(const float4*)(glb_rot + idx * 4);
    float ow = qw * r.x - qx * r.y - qy * r.z - qz * r.w;
    float ox = qw * r.y + qx * r.x + qy * r.w - qz * r.z;
    float oy = qw * r.z - qx * r.w + qy * r.x + qz * r.y;
    float oz = qw * r.w + qx * r.z - qy * r.y + qz * r.x;
    if (ow < 0.0f) { ow = -ow; ox = -ox; oy = -oy; oz = -oz; }
    float4 o; o.x = ow; o.y = ox; o.z = oy; o.w = oz;
    *(float4*)(out_rot + idx * 4) = o;
}

// ---------------------------------------------------------------------------
// Host entry
// ---------------------------------------------------------------------------
extern "C" void kernel_launch(void* const* d_in, const int* in_sizes, int n_in,
                              void* d_out, int out_size, void* d_ws, size_t ws_size,
                              hipStream_t stream) {
    const float* glb_pos = (const float*)d_in[0];   // [B,T,J,3]
    const float* glb_rot = (const float*)d_in[1];   // [B,T,J,4]
    const float* glb_vel = (const float*)d_in[2];   // [B,T-1,J,3]
    const float* root    = (const float*)d_in[3];   // [B,T,1,4]

    const long long root_n = (long long)in_sizes[3] / 4;          // B*T
    const long long pos_n  = (long long)in_sizes[0] / 3;          // B*T*J
    const int J            = (int)(pos_n / root_n);
    const long long vrows  = (long long)in_sizes[2] / (3LL * J);  // B*(T-1)
    const int B            = (int)(root_n - vrows);
    const int T            = (int)(root_n / B);

    float* traj    = (float*)d_ws;                 // B*T*2 floats
    float* out_pos = (float*)d_out;                // B*T*J*3
    float* out_rot = out_pos + (size_t)pos_n * 3;  // B*T*J*4

    traj_scan_kernel<<<B, T, 0, stream>>>(glb_pos, glb_vel, root, traj, T, J);

    if (J == 32 && (root_n % APPLY_PAIRS) == 0) {
        apply32_kernel<<<(int)(root_n / APPLY_PAIRS), 256, 0, stream>>>(
            glb_pos, glb_rot, root, traj, out_pos, out_rot);
    } else {
        long long total = pos_n;
        int threads = 256;
        long long blocks = (total + threads - 1) / threads;
        apply_generic_kernel<<<(int)blocks, threads, 0, stream>>>(
            glb_pos, glb_rot, root, traj, out_pos, out_rot, total, J);
    }
}